// LambdaLayer_76252849373193
// MI455X (gfx1250) — compile-verified
//
#include <hip/hip_runtime.h>
#include <hip/hip_bf16.h>
#include <math.h>

// ---------------------------------------------------------------------------
// LambdaLayer (LambdaNetworks) for MI455X / gfx1250, fp32, wave32.
// Heavy GEMMs use V_WMMA_F32_16X16X4_F32; x staged through LDS exactly once
// (async double-buffered via GLOBAL_LOAD_ASYNC_TO_LDS_B128);
// conv part fused into final kernel; lam_c split-K over 16 partials.
// ---------------------------------------------------------------------------

typedef __attribute__((ext_vector_type(2))) float v2f;
typedef __attribute__((ext_vector_type(4))) float v4f;
typedef __attribute__((ext_vector_type(8))) float v8f;
typedef int v4i __attribute__((__vector_size__(4 * sizeof(int))));

#define BATCH   16
#define CIN     256
#define HH      64
#define WW      64
#define NPOS    4096          // HH*WW
#define HEADS   4
#define DIMK    16
#define DIMV    64
#define RKS     7
#define RTAPS   49            // 7*7
#define NCH     144           // 64 q + 16 k + 64 v
#define EPSBN   1e-5f
#define KSPLIT  16            // split-K factor for lam_c

// workspace layout (in floats)
#define OFF_QKV   0
#define QKV_ELEMS (BATCH * NCH * NPOS)          // 9,437,184
#define OFF_MEAN  (QKV_ELEMS)                   // 128 entries (64 q + 64 v)
#define OFF_RSTD  (QKV_ELEMS + 128)             // 128 entries
#define OFF_RMAX  (QKV_ELEMS + 256)             // 256 entries (b*16 + kc)
#define OFF_RSUM  (QKV_ELEMS + 512)             // 256 entries
#define OFF_LAMC  (QKV_ELEMS + 1024)            // KSPLIT*16*16*64 partials

#define AS1 __attribute__((address_space(1)))
#define AS3 __attribute__((address_space(3)))

#if __has_builtin(__builtin_amdgcn_global_load_async_to_lds_b128)
#define HAVE_ASYNC_LDS 1
#else
#define HAVE_ASYNC_LDS 0
#endif

__device__ __forceinline__ void wait_async0() {
#if __has_builtin(__builtin_amdgcn_s_wait_asynccnt)
    __builtin_amdgcn_s_wait_asynccnt(0);
#else
    asm volatile("s_wait_asynccnt 0x0" ::: "memory");
#endif
}

// ---------------------------------------------------------------------------
// Kernel 1: qkv[b, ch, n] = W[ch, :] @ x[b, :, n]   (WMMA fp32 16x16x4)
// grid = (NPOS/64, BATCH), block = 128 (4 waves, 16 n-cols each).
// x tile (256 K x 64 N) staged through LDS in two 128-row stages (double-
// buffered via GLOBAL_LOAD_ASYNC_TO_LDS when available) and reused by all
// 9 channel tiles -> x streams from HBM exactly once.
// ---------------------------------------------------------------------------
__global__ __launch_bounds__(128)
void proj_gemm_kernel(const float* __restrict__ x,
                      const float* __restrict__ Wq,
                      const float* __restrict__ Wk,
                      const float* __restrict__ Wv,
                      float* __restrict__ ws)
{
    __shared__ float xs[2 * 128 * 64];    // double-buffered K-stages (2x32 KB)

    const int tid   = threadIdx.x;
    const int wave  = tid >> 5;
    const int lane  = tid & 31;
    const int lo    = lane & 15;          // A row (M) / D col (N) selector
    const int hi    = lane >> 4;          // 0: K 0..1 / M 0..7 ; 1: K 2..3 / M 8..15
    const int n0    = blockIdx.x * 64;    // block n-tile base
    const int nw    = wave * 16;          // wave n-subtile within the block
    const int b     = blockIdx.y;
    const int koff  = hi * 2;

    const float* xb = x + (size_t)b * CIN * NPOS + n0;

    v8f acc[9];
    #pragma unroll
    for (int t = 0; t < 9; ++t) acc[t] = (v8f){};

#if HAVE_ASYNC_LDS
    // issue async staging of one 128x64 K-stage into buffer `buf`
    auto stage_async = [&](int stage, int buf) {
        float* dst = xs + buf * (128 * 64);
        for (int j = tid; j < (128 * 64) / 4; j += 128) {
            int Kr = j >> 4;
            int nn = (j & 15) * 4;
            __builtin_amdgcn_global_load_async_to_lds_b128(
                (AS1 v4i*)(xb + (size_t)(stage * 128 + Kr) * NPOS + nn),
                (AS3 v4i*)(dst + Kr * 64 + nn),
                0, 0);
        }
    };
    stage_async(0, 0);
    wait_async0();
    __syncthreads();
    stage_async(1, 1);                    // overlap stage-1 copy with stage-0 math
#else
    {   // synchronous fallback staging of stage 0
        float* dst = xs;
        for (int j = tid; j < (128 * 64) / 4; j += 128) {
            int Kr = j >> 4;
            int nn = (j & 15) * 4;
            *(v4f*)(dst + Kr * 64 + nn) =
                *(const v4f*)(xb + (size_t)Kr * NPOS + nn);
        }
        __syncthreads();
    }
#endif

    #pragma unroll
    for (int stage = 0; stage < 2; ++stage) {
        const float* xsb = xs + stage * (128 * 64);

        #pragma unroll
        for (int t = 0; t < 9; ++t) {
            const float* Wrow;
            if (t < 4)       Wrow = Wq + (t * 16 + lo) * CIN;
            else if (t == 4) Wrow = Wk + lo * CIN;
            else             Wrow = Wv + ((t - 5) * 16 + lo) * CIN;

            #pragma unroll 4
            for (int kk = 0; kk < 128; kk += 4) {
                // A: W[ch = t*16+lo][stage*128 + kk + koff + {0,1}] (contiguous)
                v2f a = *(const v2f*)(Wrow + stage * 128 + kk + koff);
                // B: x[K = kk+koff+{0,1}][N = nw+lo] from LDS
                v2f bf;
                bf.x = xsb[(kk + koff)     * 64 + nw + lo];
                bf.y = xsb[(kk + koff + 1) * 64 + nw + lo];
                acc[t] = __builtin_amdgcn_wmma_f32_16x16x4_f32(
                             false, a, false, bf, (short)0, acc[t], false, false);
            }
        }

        if (stage == 0) {
#if HAVE_ASYNC_LDS
            wait_async0();
            __syncthreads();
#else
            __syncthreads();              // stage-0 readers done
            float* dst = xs + 128 * 64;
            for (int j = tid; j < (128 * 64) / 4; j += 128) {
                int Kr = j >> 4;
                int nn = (j & 15) * 4;
                *(v4f*)(dst + Kr * 64 + nn) =
                    *(const v4f*)(xb + (size_t)(128 + Kr) * NPOS + nn);
            }
            __syncthreads();
#endif
        }
    }

    float* qkv = ws + OFF_QKV + (size_t)b * NCH * NPOS;
    #pragma unroll
    for (int t = 0; t < 9; ++t) {
        #pragma unroll
        for (int r = 0; r < 8; ++r) {
            int chd = t * 16 + r + hi * 8;                 // D: vgpr r -> M=r / r+8
            qkv[(size_t)chd * NPOS + n0 + nw + lo] = acc[t][r];
        }
    }
}

// ---------------------------------------------------------------------------
// Kernel 2: BN stats (mean, rstd) per channel over b*n for q and v channels.
// grid = 128 (64 q + 64 v), block = 256
// ---------------------------------------------------------------------------
__global__ __launch_bounds__(256)
void bn_stats_kernel(float* __restrict__ ws)
{
    __shared__ float s1[256];
    __shared__ float s2[256];
    const int cid = blockIdx.x;                       // 0..127
    const int qch = (cid < 64) ? cid : (80 + (cid - 64));
    const float* qkv = ws + OFF_QKV;

    float a = 0.f, b2 = 0.f;
    for (int i = threadIdx.x; i < BATCH * NPOS; i += 256) {
        int bb = i >> 12;
        int n  = i & (NPOS - 1);
        float v = qkv[((size_t)bb * NCH + qch) * NPOS + n];
        a  += v;
        b2 += v * v;
    }
    s1[threadIdx.x] = a;
    s2[threadIdx.x] = b2;
    __syncthreads();
    for (int off = 128; off > 0; off >>= 1) {
        if (threadIdx.x < off) {
            s1[threadIdx.x] += s1[threadIdx.x + off];
            s2[threadIdx.x] += s2[threadIdx.x + off];
        }
        __syncthreads();
    }
    if (threadIdx.x == 0) {
        const float inv = 1.0f / (BATCH * NPOS);
        float mean = s1[0] * inv;
        float var  = s2[0] * inv - mean * mean;
        ws[OFF_MEAN + cid] = mean;
        ws[OFF_RSTD + cid] = rsqrtf(var + EPSBN);
    }
}

// ---------------------------------------------------------------------------
// Kernel 3: softmax stats (row max, sum of exp) per (b, k-channel) over n.
// grid = 256 (b*16 + kc), block = 256
// ---------------------------------------------------------------------------
__global__ __launch_bounds__(256)
void softmax_stats_kernel(float* __restrict__ ws)
{
    __shared__ float sm[256];
    const int rid = blockIdx.x;                       // b*16 + kc
    const int b   = rid >> 4;
    const int kc  = rid & 15;
    const float* row = ws + OFF_QKV + ((size_t)b * NCH + 64 + kc) * NPOS;

    float m = -3.4e38f;
    for (int i = threadIdx.x; i < NPOS; i += 256)
        m = fmaxf(m, row[i]);
    sm[threadIdx.x] = m;
    __syncthreads();
    for (int off = 128; off > 0; off >>= 1) {
        if (threadIdx.x < off)
            sm[threadIdx.x] = fmaxf(sm[threadIdx.x], sm[threadIdx.x + off]);
        __syncthreads();
    }
    const float rmax = sm[0];
    __syncthreads();

    float s = 0.f;
    for (int i = threadIdx.x; i < NPOS; i += 256)
        s += expf(row[i] - rmax);
    sm[threadIdx.x] = s;
    __syncthreads();
    for (int off = 128; off > 0; off >>= 1) {
        if (threadIdx.x < off)
            sm[threadIdx.x] += sm[threadIdx.x + off];
        __syncthreads();
    }
    if (threadIdx.x == 0) {
        ws[OFF_RMAX + rid] = rmax;
        ws[OFF_RSUM + rid] = sm[0];
    }
}

// ---------------------------------------------------------------------------
// Kernel 4: apply BN (q, v channels) and softmax (k channels) in place.
// grid = QKV_ELEMS/256, block = 256
// ---------------------------------------------------------------------------
__global__ __launch_bounds__(256)
void normalize_kernel(const float* __restrict__ gamma_q,
                      const float* __restrict__ beta_q,
                      const float* __restrict__ gamma_v,
                      const float* __restrict__ beta_v,
                      float* __restrict__ ws)
{
    size_t idx = (size_t)blockIdx.x * 256 + threadIdx.x;
    if (idx >= (size_t)QKV_ELEMS) return;
    int b   = (int)(idx / ((size_t)NCH * NPOS));
    int rem = (int)(idx - (size_t)b * NCH * NPOS);
    int ch  = rem >> 12;                               // NPOS = 4096
    float v = ws[OFF_QKV + idx];
    if (ch < 64) {
        float mean = ws[OFF_MEAN + ch];
        float rstd = ws[OFF_RSTD + ch];
        v = (v - mean) * rstd * gamma_q[ch] + beta_q[ch];
    } else if (ch < 80) {
        int rid = b * 16 + (ch - 64);
        v = expf(v - ws[OFF_RMAX + rid]) / ws[OFF_RSUM + rid];
    } else {
        int vc = ch - 80;
        float mean = ws[OFF_MEAN + 64 + vc];
        float rstd = ws[OFF_RSTD + 64 + vc];
        v = (v - mean) * rstd * gamma_v[vc] + beta_v[vc];
    }
    ws[OFF_QKV + idx] = v;
}

// ---------------------------------------------------------------------------
// Kernel 5: lam_c partials (split-K over n):
//   lamc_part[ks][b,k,v] = sum_{n in ks-chunk} softk[b,k,n] * vbn[b,v,n]
// grid = (4 v-tiles, BATCH, KSPLIT), block = 32 (one wave, WMMA K=256 chunk)
// ---------------------------------------------------------------------------
__global__ __launch_bounds__(32)
void lamc_gemm_kernel(float* __restrict__ ws)
{
    const int lane = threadIdx.x & 31;
    const int lo   = lane & 15;
    const int hi   = lane >> 4;
    const int vt   = blockIdx.x;                      // v tile (16 wide)
    const int b    = blockIdx.y;
    const int ks   = blockIdx.z;                      // K-split chunk
    const int koff = hi * 2;
    const int nbeg = ks * (NPOS / KSPLIT);            // 256-wide chunk

    const float* kbase = ws + OFF_QKV + ((size_t)b * NCH + 64 + lo) * NPOS + nbeg;
    const float* vbase = ws + OFF_QKV + ((size_t)b * NCH + 80 + vt * 16 + lo) * NPOS + nbeg;

    v8f acc = {};
    #pragma unroll 8
    for (int n = 0; n < NPOS / KSPLIT; n += 4) {
        // A[m = k-ch = lo][K = n + koff + {0,1}] — contiguous along n
        v2f a = *(const v2f*)(kbase + n + koff);
        // B[K = n + koff + {0,1}][N = v-ch = lo] — also contiguous along n
        v2f bf = *(const v2f*)(vbase + n + koff);
        acc = __builtin_amdgcn_wmma_f32_16x16x4_f32(
                  false, a, false, bf, (short)0, acc, false, false);
    }

    float* lamc = ws + OFF_LAMC + ((size_t)ks * BATCH + b) * (DIMK * DIMV);
    #pragma unroll
    for (int r = 0; r < 8; ++r) {
        int kch = r + hi * 8;
        lamc[kch * DIMV + vt * 16 + lo] = acc[r];
    }
}

// ---------------------------------------------------------------------------
// Kernel 6: fused local conv (lam_p) + contraction over k.
// out[b, h*64+v, n] = sum_k q[b,h*16+k,n] * (lam_c[k,v] + sum_t R[k,t]*vpatch)
// grid = (HH/2 row-pairs, BATCH), block = 128 (two image rows per block)
// ---------------------------------------------------------------------------
__global__ __launch_bounds__(128)
void fused_out_kernel(const float* __restrict__ R,
                      const float* __restrict__ ws,
                      float* __restrict__ out)
{
    __shared__ float qs[64 * 128];       // [h*16+k][row-pair*64 + w] (32 KB)
    __shared__ float Rs[DIMK * RTAPS];   // 16 x 49
    __shared__ float lcs[DIMK * DIMV];   // 16 x 64 (sum of split-K partials)

    const int rowbase = blockIdx.x * 2;
    const int b   = blockIdx.y;
    const int tid = threadIdx.x;
    const float* qkv = ws + OFF_QKV + (size_t)b * NCH * NPOS;

    for (int i = tid; i < 64 * 128; i += 128) {
        int hk  = i >> 7;
        int rem = i & 127;                             // 2 rows x 64 cols
        qs[i] = qkv[(size_t)hk * NPOS + rowbase * WW + rem];
    }
    for (int i = tid; i < DIMK * RTAPS; i += 128) Rs[i] = R[i];
    for (int i = tid; i < DIMK * DIMV; i += 128) {
        float s = 0.f;
        #pragma unroll
        for (int ks = 0; ks < KSPLIT; ++ks)
            s += ws[OFF_LAMC + ((size_t)ks * BATCH + b) * (DIMK * DIMV) + i];
        lcs[i] = s;
    }
    __syncthreads();

    const int row = rowbase + (tid >> 6);
    const int w   = tid & 63;
    const int col = tid & 127;                         // index into qs row-pair

    for (int v = 0; v < DIMV; ++v) {
        const float* vb = qkv + (size_t)(80 + v) * NPOS;
        float patch[RTAPS];
        int p = 0;
        #pragma unroll
        for (int dy = -3; dy <= 3; ++dy) {
            int r2 = row + dy;
            #pragma unroll
            for (int dx = -3; dx <= 3; ++dx) {
                int c2 = w + dx;
                patch[p++] = (r2 >= 0 && r2 < HH && c2 >= 0 && c2 < WW)
                                 ? vb[r2 * WW + c2] : 0.0f;
            }
        }
        float a0 = 0.f, a1 = 0.f, a2 = 0.f, a3 = 0.f;
        for (int k = 0; k < DIMK; ++k) {
            float s = lcs[k * DIMV + v];
            #pragma unroll
            for (int t = 0; t < RTAPS; ++t)
                s = fmaf(Rs[k * RTAPS + t], patch[t], s);
            a0 = fmaf(qs[(0 * DIMK + k) * 128 + col], s, a0);
            a1 = fmaf(qs[(1 * DIMK + k) * 128 + col], s, a1);
            a2 = fmaf(qs[(2 * DIMK + k) * 128 + col], s, a2);
            a3 = fmaf(qs[(3 * DIMK + k) * 128 + col], s, a3);
        }
        const size_t nidx = (size_t)row * WW + w;
        float* ob = out + (size_t)b * (HEADS * DIMV) * NPOS + nidx;
        ob[(size_t)(0 * DIMV + v) * NPOS] = a0;
        ob[(size_t)(1 * DIMV + v) * NPOS] = a1;
        ob[(size_t)(2 * DIMV + v) * NPOS] = a2;
        ob[(size_t)(3 * DIMV + v) * NPOS] = a3;
    }
}

// ---------------------------------------------------------------------------
// Host-side launcher
// ---------------------------------------------------------------------------
extern "C" void kernel_launch(void* const* d_in, const int* in_sizes, int n_in,
                              void* d_out, int out_size, void* d_ws, size_t ws_size,
                              hipStream_t stream)
{
    const float* x       = (const float*)d_in[0];
    const float* Wq      = (const float*)d_in[1];
    const float* Wk      = (const float*)d_in[2];
    const float* Wv      = (const float*)d_in[3];
    const float* gamma_q = (const float*)d_in[4];
    const float* beta_q  = (const float*)d_in[5];
    const float* gamma_v = (const float*)d_in[6];
    const float* beta_v  = (const float*)d_in[7];
    const float* R       = (const float*)d_in[8];
    float* out = (float*)d_out;
    float* ws  = (float*)d_ws;

    // 1) projections (WMMA fp32, x async-staged through LDS once)
    proj_gemm_kernel<<<dim3(NPOS / 64, BATCH), 128, 0, stream>>>(
        x, Wq, Wk, Wv, ws);
    // 2) BN stats
    bn_stats_kernel<<<128, 256, 0, stream>>>(ws);
    // 3) softmax stats
    softmax_stats_kernel<<<256, 256, 0, stream>>>(ws);
    // 4) normalize in place
    normalize_kernel<<<(QKV_ELEMS + 255) / 256, 256, 0, stream>>>(
        gamma_q, beta_q, gamma_v, beta_v, ws);
    // 5) lam_c partials (WMMA fp32, split-K x16)
    lamc_gemm_kernel<<<dim3(DIMV / 16, BATCH, KSPLIT), 32, 0, stream>>>(ws);
    // 6) fused conv + contraction (sums lam_c partials)
    fused_out_kernel<<<dim3(HH / 2, BATCH), 128, 0, stream>>>(R, ws, out);
}